// TtLlamaQKV_optimized_86131274154272
// MI455X (gfx1250) — compile-verified
//
#include <hip/hip_runtime.h>
#include <hip/hip_bf16.h>

typedef __attribute__((ext_vector_type(2))) float v2f;
typedef __attribute__((ext_vector_type(4))) float v4f;
typedef __attribute__((ext_vector_type(8))) float v8f;

#define DIM      8192
#define BATCH    32
#define E_Q      8192   // N_HEADS * HEAD_DIM
#define E_KV     1024   // N_KV_HEADS * HEAD_DIM
#define Q_ELEMS  (E_Q  * BATCH)   // 262144
#define KV_ELEMS (E_KV * BATCH)   // 32768
#define PF_DIST  1024   // prefetch 4 KB ahead per weight row

// One wave computes a 16(E) x 16(B) fp32 output tile via V_WMMA_F32_16X16X4_F32,
// looping K=8192 in chunks of 32 (8 WMMAs + 8 A-loads + 8 B-loads per chunk).
// Block = 4 waves = 2 E-tiles x 2 batch-halves (the two batch-half waves share
// the same weight rows -> WGP$/L2 hit, weights stream from HBM exactly once).
__global__ __launch_bounds__(128) void qkv_wmma_f32_kernel(
    const float* __restrict__ x,
    const float* __restrict__ wq,
    const float* __restrict__ wk,
    const float* __restrict__ wv,
    float* __restrict__ out)
{
    const int lane  = threadIdx.x & 31;
    const int lmod  = lane & 15;      // N (batch) index / A row index
    const int lhalf = lane >> 4;      // K phase {0,1} vs {2,3}; M+8 on output

    // Wave-uniform tile coordinates, forced into SGPRs so the wq/wk/wv region
    // select compiles to scalar branches instead of an exec-mask cascade.
    const int wave  = __builtin_amdgcn_readfirstlane(threadIdx.x >> 5);
    const int etile = blockIdx.x * 2 + (wave >> 1);   // 0..639
    const int n0    = (wave & 1) * 16;                // batch half
    const int e0    = etile * 16;                     // global feature row

    const float* w;
    float* obase;
    int eo;
    if (e0 < E_Q)              { w = wq; obase = out;                       eo = e0; }
    else if (e0 < E_Q + E_KV)  { w = wk; obase = out + Q_ELEMS;             eo = e0 - E_Q; }
    else                       { w = wv; obase = out + Q_ELEMS + KV_ELEMS;  eo = e0 - (E_Q + E_KV); }

    // A operand (weights): lane holds row (eo + lmod), K pairs starting at 2*lhalf.
    const float* __restrict__ pa = w + (size_t)(eo + lmod) * DIM + 2 * lhalf;
    // B operand (x): lane holds batch column (n0 + lmod), same K phase.
    const float* __restrict__ pb = x + (size_t)(n0 + lmod) * DIM + 2 * lhalf;

    v8f acc = {};   // C/D accumulator: 16x16 f32, 8 VGPRs per lane

    for (int k0 = 0; k0 < DIM; k0 += 32) {
        // Stream-prefetch the weight rows 4 KB ahead (scalar-uniform guard).
        // 16 distinct cache lines per instruction; each line prefetched once.
        if (k0 + PF_DIST < DIM)
            __builtin_prefetch(pa + k0 + PF_DIST, 0, 0);  // global_prefetch_b8

        v2f a[8], b[8];
        #pragma unroll
        for (int s = 0; s < 8; ++s) {
            a[s] = *(const v2f*)(pa + k0 + s * 4);   // {K+2h, K+2h+1} of weight rows
            b[s] = *(const v2f*)(pb + k0 + s * 4);   // {K+2h, K+2h+1} of x columns
        }
        #pragma unroll
        for (int s = 0; s < 8; ++s) {
            // 8 args: (neg_a, A, neg_b, B, c_mod, C, reuse_a, reuse_b)
            acc = __builtin_amdgcn_wmma_f32_16x16x4_f32(
                false, a[s], false, b[s], (short)0, acc, false, false);
        }
    }

    // Store: VGPR r holds (M = r + 8*lhalf, N = lmod). For a fixed lane the 8
    // accumulator regs are 8 consecutive dh positions (a 16-row tile never
    // crosses a 128-wide head). Region layout: out[h*B*128 + b*128 + dh].
    const int h  = eo >> 7;
    const int dh = (eo & 127) + 8 * lhalf;
    float* po = obase + (size_t)h * (BATCH * 128)
                      + (size_t)(n0 + lmod) * 128
                      + dh;

    v4f lo = { acc.s0, acc.s1, acc.s2, acc.s3 };
    v4f hi = { acc.s4, acc.s5, acc.s6, acc.s7 };
    *(v4f*)(po)     = lo;   // global_store_b128
    *(v4f*)(po + 4) = hi;   // global_store_b128
}

extern "C" void kernel_launch(void* const* d_in, const int* in_sizes, int n_in,
                              void* d_out, int out_size, void* d_ws, size_t ws_size,
                              hipStream_t stream) {
    const float* x  = (const float*)d_in[0];
    const float* wq = (const float*)d_in[1];
    const float* wk = (const float*)d_in[2];
    const float* wv = (const float*)d_in[3];
    float* out = (float*)d_out;

    // 640 E-tiles total (10240/16); 2 E-tiles per block -> 320 blocks of 128 thr.
    dim3 grid(320, 1, 1);
    dim3 block(128, 1, 1);
    qkv_wmma_f32_kernel<<<grid, block, 0, stream>>>(x, wq, wk, wv, out);
}